// DynamicFilterChannel0_65721589563905
// MI455X (gfx1250) — compile-verified
//
#include <hip/hip_runtime.h>
#include <hip/hip_bf16.h>
#include <math.h>

typedef __attribute__((ext_vector_type(2))) float v2f;
typedef __attribute__((ext_vector_type(4))) float v4f;
typedef __attribute__((ext_vector_type(8))) float v8f;

#define N_      16
#define C_      256
#define H_      128
#define W_      128
#define GK2     72      // GROUP * K * K
#define CPG     32      // channels per group (256/8)
#define HW_     (H_ * W_)
#define TROWS   16
#define LDSW    136     // padded LDS row: [0..2] pad, [3] left halo, [4..131] data, [132] right halo

// ---------------------------------------------------------------------------
// Kernel 1: global average pool over H*W for each (n,c). One block per plane.
// ---------------------------------------------------------------------------
__global__ void dfc_pool_kernel(const float* __restrict__ x,
                                float* __restrict__ pooled) {
    const int nc = blockIdx.x;                       // 0 .. N*C-1
    const v4f* p4 = (const v4f*)(x + (size_t)nc * HW_);
    float s = 0.0f;
    for (int i = threadIdx.x; i < HW_ / 4; i += 256) {   // 16 coalesced b128 loads
        v4f v = p4[i];
        s += (v.x + v.y) + (v.z + v.w);
    }
    __shared__ float red[256];
    red[threadIdx.x] = s;
    __syncthreads();
    for (int off = 128; off > 0; off >>= 1) {
        if (threadIdx.x < off) red[threadIdx.x] += red[threadIdx.x + off];
        __syncthreads();
    }
    if (threadIdx.x == 0) pooled[nc] = red[0] * (1.0f / (float)HW_);
}

// ---------------------------------------------------------------------------
// Kernel 2: whole weight pipeline in one block (5 waves = 160 threads).
//   lf = pooled @ conv_w^T   (16x256 @ 256x72) -> v_wmma_f32_16x16x4_f32
//   z  = lf @ gate_w^T       (16x72  @ 72x72)  -> v_wmma_f32_16x16x4_f32
//   lf = lf * sigmoid(z); BN; softmax over 9 taps/group -> wk (16x72)
// Out-of-range output columns (72..79) use a clamped B row + select-zero so
// the WMMA loop has no exec-mask branches (WMMA requires EXEC all-ones).
// ---------------------------------------------------------------------------
__global__ void dfc_weights_kernel(const float* __restrict__ pooled,
                                   const float* __restrict__ conv_w,
                                   const float* __restrict__ gate_w,
                                   const float* __restrict__ bn_gamma,
                                   const float* __restrict__ bn_beta,
                                   const float* __restrict__ bn_mean,
                                   const float* __restrict__ bn_var,
                                   float* __restrict__ wk) {
    __shared__ float sp[N_ * C_];       // pooled staged in LDS (16 KB)
    __shared__ float lf[N_][80];        // conv output, padded to 80 cols
    __shared__ float gated[N_][80];     // post-gate/BN

    const int tid  = threadIdx.x;       // 0..159
    const int wave = tid >> 5;          // 0..4 : N-tile of the 72/80 outputs
    const int lane = tid & 31;
    const int lm   = lane & 15;         // m (A rows) / n-in-tile (B,C cols)
    const int lh   = lane >> 4;         // lane half -> K sub-block
    const int ncol = wave * 16 + lm;    // global output column (0..79)
    const bool bvalid = (ncol < GK2);
    const int  bcol   = bvalid ? ncol : 0;   // clamped, always-valid B row

    for (int i = tid; i < N_ * C_; i += 160) sp[i] = pooled[i];
    __syncthreads();

    // ---- conv: lf = pooled @ conv_w^T, K = 256 (64 WMMA steps) ----
    const float* bc = conv_w + bcol * C_;
    v8f acc = {0.f, 0.f, 0.f, 0.f, 0.f, 0.f, 0.f, 0.f};
    for (int k0 = 0; k0 < C_; k0 += 4) {
        const int ka = k0 + lh * 2;
        v2f a, b;
        a.x = sp[lm * C_ + ka];
        a.y = sp[lm * C_ + ka + 1];
        const float bx = bc[ka], by = bc[ka + 1];   // unconditional load
        b.x = bvalid ? bx : 0.f;                    // v_cndmask, no exec branch
        b.y = bvalid ? by : 0.f;
        acc = __builtin_amdgcn_wmma_f32_16x16x4_f32(
            false, a, false, b, (short)0, acc, false, false);
    }
    for (int r = 0; r < 8; ++r) {
        lf[r + 8 * lh][wave * 16 + lm] = acc[r];
    }
    __syncthreads();

    // ---- gate: z = lf @ gate_w^T, K = 72 (18 WMMA steps) ----
    const float* bg = gate_w + bcol * GK2;
    v8f accz = {0.f, 0.f, 0.f, 0.f, 0.f, 0.f, 0.f, 0.f};
    for (int k0 = 0; k0 < GK2; k0 += 4) {
        const int ka = k0 + lh * 2;
        v2f a, b;
        a.x = lf[lm][ka];
        a.y = lf[lm][ka + 1];
        const float bx = bg[ka], by = bg[ka + 1];
        b.x = bvalid ? bx : 0.f;
        b.y = bvalid ? by : 0.f;
        accz = __builtin_amdgcn_wmma_f32_16x16x4_f32(
            false, a, false, b, (short)0, accz, false, false);
    }

    // ---- sigmoid gate + BN (eval) ----
    if (bvalid) {
        const float inv_std = bn_gamma[ncol] * rsqrtf(bn_var[ncol] + 1e-5f);
        const float mu      = bn_mean[ncol];
        const float bb      = bn_beta[ncol];
        for (int r = 0; r < 8; ++r) {
            const int m = r + 8 * lh;
            const float l = lf[m][ncol];
            const float sg = 1.0f / (1.0f + __expf(-accz[r]));
            gated[m][ncol] = (l * sg - mu) * inv_std + bb;
        }
    }
    __syncthreads();

    // ---- softmax over 9 taps per (n, group): 16*8 = 128 jobs ----
    if (tid < N_ * 8) {
        const int m = tid >> 3;
        const int g = tid & 7;
        float v[9];
        float mx = -3.0e38f;
        for (int t = 0; t < 9; ++t) {
            v[t] = gated[m][g * 9 + t];
            mx = fmaxf(mx, v[t]);
        }
        float sum = 0.f;
        for (int t = 0; t < 9; ++t) { v[t] = __expf(v[t] - mx); sum += v[t]; }
        const float inv = 1.0f / sum;
        for (int t = 0; t < 9; ++t) wk[m * GK2 + g * 9 + t] = v[t] * inv;
    }
}

// ---------------------------------------------------------------------------
// Kernel 3: reflect-padded 3x3 weighted average + residual.
// Block = 128 threads = (32 float4-columns) x (4 row groups). Each block does
// a 16-row strip of one (n,c) plane:
//   - fill 18 x (128+halo) LDS tile with global_load_b128 (reflection at load)
//   - each thread computes 4 adjacent columns x 4 rows via ds_load_b128 + 2
//     halo b32s per stencil row
//   - emit low / high with NON-TEMPORAL global_store_b128 (write-once streams,
//     don't thrash L2 so x halo rows stay resident across strips)
// ---------------------------------------------------------------------------
__global__ void dfc_filter_kernel(const float* __restrict__ x,
                                  const float* __restrict__ wk,
                                  float* __restrict__ low,
                                  float* __restrict__ high) {
    const int strip = blockIdx.x;          // 0..H/TROWS-1
    const int nc    = blockIdx.y;          // 0..N*C-1
    const int n     = nc >> 8;             // / C_
    const int c     = nc & (C_ - 1);
    const int g     = c / CPG;             // group index

    __shared__ float tile[TROWS + 2][LDSW];
    __shared__ float w9[9];

    const int tid = threadIdx.x;
    const int tx  = tid & 31;              // float4 column group (0..31)
    const int ty  = tid >> 5;              // row group (0..3)
    if (tid < 9) w9[tid] = wk[n * GK2 + g * 9 + tid];

    const size_t base = (size_t)nc * HW_;
    const float* xp = x + base;
    const int y0 = strip * TROWS;

    for (int r = ty; r < TROWS + 2; r += 4) {
        int iy = y0 + r - 1;
        iy = (iy < 0) ? -iy : ((iy >= H_) ? (2 * H_ - 2 - iy) : iy);
        const float* row = xp + (size_t)iy * W_;
        v4f v = *(const v4f*)(row + 4 * tx);              // global_load_b128
        *(v4f*)&tile[r][4 * tx + 4] = v;                  // 16B-aligned ds_store
        if (tx == 0)  tile[r][3]   = row[1];              // reflect col -1 -> 1
        if (tx == 31) tile[r][132] = row[W_ - 2];         // reflect col W -> W-2
    }
    __syncthreads();

    float wr9[9];
    #pragma unroll
    for (int t = 0; t < 9; ++t) wr9[t] = w9[t];

    #pragma unroll
    for (int k = 0; k < TROWS / 4; ++k) {
        const int r = ty + 4 * k;          // output row within strip
        v4f lo = {0.f, 0.f, 0.f, 0.f};
        v4f mid = {0.f, 0.f, 0.f, 0.f};
        #pragma unroll
        for (int i = 0; i < 3; ++i) {
            const float* trow = &tile[r + i][4 * tx];
            const float lft = trow[3];                    // left halo tap
            const v4f   m   = *(const v4f*)(trow + 4);    // ds_load_b128
            const float rgt = trow[8];                    // right halo tap
            const float wl = wr9[3 * i], wm = wr9[3 * i + 1], wrr = wr9[3 * i + 2];
            lo.x += wl * lft + wm * m.x + wrr * m.y;
            lo.y += wl * m.x + wm * m.y + wrr * m.z;
            lo.z += wl * m.y + wm * m.z + wrr * m.w;
            lo.w += wl * m.z + wm * m.w + wrr * rgt;
            if (i == 1) mid = m;
        }
        v4f hi;
        hi.x = mid.x - lo.x; hi.y = mid.y - lo.y;
        hi.z = mid.z - lo.z; hi.w = mid.w - lo.w;
        const size_t idx = base + (size_t)(y0 + r) * W_ + 4 * tx;
        __builtin_nontemporal_store(lo, (v4f*)(low + idx));   // b128 NT stores
        __builtin_nontemporal_store(hi, (v4f*)(high + idx));
    }
}

// ---------------------------------------------------------------------------
extern "C" void kernel_launch(void* const* d_in, const int* in_sizes, int n_in,
                              void* d_out, int out_size, void* d_ws, size_t ws_size,
                              hipStream_t stream) {
    const float* x        = (const float*)d_in[0];
    const float* conv_w   = (const float*)d_in[1];
    const float* gate_w   = (const float*)d_in[2];
    const float* bn_gamma = (const float*)d_in[3];
    const float* bn_beta  = (const float*)d_in[4];
    const float* bn_mean  = (const float*)d_in[5];
    const float* bn_var   = (const float*)d_in[6];

    float* out  = (float*)d_out;
    float* lowp = out;                                   // (n,c,h,w)
    float* high = out + (size_t)N_ * C_ * HW_;           // (n,c,h,w)

    float* pooled = (float*)d_ws;                        // 16*256 floats
    float* wk     = pooled + N_ * C_;                    // 16*72 floats

    dfc_pool_kernel<<<N_ * C_, 256, 0, stream>>>(x, pooled);

    dfc_weights_kernel<<<1, 160, 0, stream>>>(pooled, conv_w, gate_w,
                                              bn_gamma, bn_beta, bn_mean, bn_var,
                                              wk);

    dim3 grid(H_ / TROWS, N_ * C_);
    dfc_filter_kernel<<<grid, 128, 0, stream>>>(x, wk, lowp, high);
}